// Net_68298569941027
// MI455X (gfx1250) — compile-verified
//
#include <hip/hip_runtime.h>
#include <hip/hip_bf16.h>

typedef float v2f __attribute__((ext_vector_type(2)));
typedef float v8f __attribute__((ext_vector_type(8)));

#define F_IN 512
#define HID  64
#define NCLS 40

// ---------------- degree accumulation (indeg via atomics) ----------------
__global__ void k_deg(const int* __restrict__ dst, float* __restrict__ deg, unsigned E) {
  unsigned i = blockIdx.x * blockDim.x + threadIdx.x;
  if (i < E) atomicAdd(&deg[dst[i]], 1.0f);
}

// deg_inv_sqrt in place: dis[i] = rsqrt(indeg + 1)   (self-loop)
__global__ void k_dis(float* __restrict__ dis, unsigned N) {
  unsigned i = blockIdx.x * blockDim.x + threadIdx.x;
  if (i < N) dis[i] = rsqrtf(dis[i] + 1.0f);
}

// ---------------- GEMM1: H = X[100000x512] * W1[512x64], fp32 WMMA ----------------
// one wave per 16-row tile, 4 N-tiles of 16 held in 4 accumulators
__global__ void k_gemm1(const float* __restrict__ X, const float* __restrict__ W,
                        float* __restrict__ H, int mtiles) {
  int wave = threadIdx.x >> 5;
  int lane = threadIdx.x & 31;
  int hlf  = lane >> 4;      // lane half selects K-pair / M-half per ISA layout
  int idx  = lane & 15;
  int tile = blockIdx.x * 4 + wave;
  if (tile >= mtiles) return;              // wave-uniform: EXEC stays all-ones
  int row0 = tile * 16;

  const float* xr = X + (size_t)(row0 + idx) * F_IN;
  v8f a0 = {0,0,0,0,0,0,0,0}, a1 = a0, a2 = a0, a3 = a0;

  for (int k = 0; k < F_IN; k += 4) {
    int kb = k + 2 * hlf;                          // this lane-half supplies K=kb,kb+1
    v2f a;  a.x = xr[kb];  a.y = xr[kb + 1];       // A[M=idx][K=kb..kb+1]
    const float* w0 = W + (size_t)kb * HID + idx;  // B[K=kb][N=idx+16t]
    const float* w1 = w0 + HID;                    // B[K=kb+1][...]
    v2f b0 = { w0[0],  w1[0]  };
    v2f b1 = { w0[16], w1[16] };
    v2f b2 = { w0[32], w1[32] };
    v2f b3 = { w0[48], w1[48] };
    a0 = __builtin_amdgcn_wmma_f32_16x16x4_f32(false, a, false, b0, (short)0, a0, false, false);
    a1 = __builtin_amdgcn_wmma_f32_16x16x4_f32(false, a, false, b1, (short)0, a1, false, false);
    a2 = __builtin_amdgcn_wmma_f32_16x16x4_f32(false, a, false, b2, (short)0, a2, false, false);
    a3 = __builtin_amdgcn_wmma_f32_16x16x4_f32(false, a, false, b3, (short)0, a3, false, false);
  }
#pragma unroll
  for (int v = 0; v < 8; ++v) {                    // C[v] = row (v + 8*half), col idx
    float* hr = H + (size_t)(row0 + v + 8 * hlf) * HID + idx;
    hr[0] = a0[v]; hr[16] = a1[v]; hr[32] = a2[v]; hr[48] = a3[v];
  }
}

// ---------------- GEMM2: Z = H1[100000x64] * W2[64x40], fp32 WMMA (N padded to 48) ----
__global__ void k_gemm2(const float* __restrict__ Hin, const float* __restrict__ W,
                        float* __restrict__ Z, int mtiles) {
  int wave = threadIdx.x >> 5;
  int lane = threadIdx.x & 31;
  int hlf  = lane >> 4;
  int idx  = lane & 15;
  int tile = blockIdx.x * 4 + wave;
  if (tile >= mtiles) return;
  int row0 = tile * 16;

  const float* xr = Hin + (size_t)(row0 + idx) * HID;
  v8f a0 = {0,0,0,0,0,0,0,0}, a1 = a0, a2 = a0;
  bool t2ok = (idx < 8);                            // cols 32..39 valid, 40..47 padded

  for (int k = 0; k < HID; k += 4) {
    int kb = k + 2 * hlf;
    v2f a;  a.x = xr[kb];  a.y = xr[kb + 1];
    const float* w0 = W + (size_t)kb * NCLS + idx;
    const float* w1 = w0 + NCLS;
    v2f b0 = { w0[0],  w1[0]  };
    v2f b1 = { w0[16], w1[16] };
    v2f b2 = { t2ok ? w0[32] : 0.0f, t2ok ? w1[32] : 0.0f };
    a0 = __builtin_amdgcn_wmma_f32_16x16x4_f32(false, a, false, b0, (short)0, a0, false, false);
    a1 = __builtin_amdgcn_wmma_f32_16x16x4_f32(false, a, false, b1, (short)0, a1, false, false);
    a2 = __builtin_amdgcn_wmma_f32_16x16x4_f32(false, a, false, b2, (short)0, a2, false, false);
  }
#pragma unroll
  for (int v = 0; v < 8; ++v) {
    float* zr = Z + (size_t)(row0 + v + 8 * hlf) * NCLS + idx;
    zr[0] = a0[v]; zr[16] = a1[v];
    if (t2ok) zr[32] = a2[v];
  }
}

// ---------------- edge scatter-add, layer 1 (64 feats) -------------------
__global__ void k_scatter1(const int* __restrict__ src, const int* __restrict__ dst,
                           const float* __restrict__ dis, const float* __restrict__ H,
                           float* __restrict__ AGG, unsigned total) {
  unsigned i = blockIdx.x * blockDim.x + threadIdx.x;
  if (i >= total) return;
  unsigned e = i >> 6, f = i & 63u;
  int s = src[e], d = dst[e];
  float nrm = dis[s] * dis[d];
  atomicAdd(&AGG[(size_t)d * HID + f], H[(size_t)s * HID + f] * nrm);
}

// finalize layer 1: add self-loop term + bias, ReLU, in place over AGG
__global__ void k_fin1(float* __restrict__ AGG, const float* __restrict__ H,
                       const float* __restrict__ dis, const float* __restrict__ b,
                       unsigned total) {
  unsigned i = blockIdx.x * blockDim.x + threadIdx.x;
  if (i >= total) return;
  unsigned r = i >> 6, f = i & 63u;
  float ds = dis[r];
  float v  = AGG[i] + ds * ds * H[i] + b[f];
  AGG[i] = v > 0.0f ? v : 0.0f;
}

// ---------------- edge scatter-add, layer 2 (40 feats, into d_out x_out half) ---------
__global__ void k_scatter2(const int* __restrict__ src, const int* __restrict__ dst,
                           const float* __restrict__ dis, const float* __restrict__ Z,
                           float* __restrict__ AGG, unsigned total) {
  unsigned i = blockIdx.x * blockDim.x + threadIdx.x;
  if (i >= total) return;
  unsigned e = i / 40u, f = i - e * 40u;
  int s = src[e], d = dst[e];
  float nrm = dis[s] * dis[d];
  atomicAdd(&AGG[(size_t)d * NCLS + f], Z[(size_t)s * NCLS + f] * nrm);
}

// finalize layer 2: self-loop + bias, in place over accumulated x_out
__global__ void k_fin2(float* __restrict__ XO, const float* __restrict__ Z,
                       const float* __restrict__ dis, const float* __restrict__ b,
                       unsigned total) {
  unsigned i = blockIdx.x * blockDim.x + threadIdx.x;
  if (i >= total) return;
  unsigned r = i / 40u, f = i - r * 40u;
  float ds = dis[r];
  XO[i] += ds * ds * Z[i] + b[f];
}

// ---------------- log_softmax: one wave32 per row of 40 ------------------
__global__ void k_lsm(const float* __restrict__ XO, float* __restrict__ LS, unsigned N) {
  unsigned row = blockIdx.x * (blockDim.x >> 5) + (threadIdx.x >> 5);
  int lane = threadIdx.x & 31;
  if (row >= N) return;
  const float* xr = XO + (size_t)row * NCLS;
  float v0 = xr[lane];
  float v1 = (lane < 8) ? xr[32 + lane] : -3.402823466e38f;
  float m = fmaxf(v0, v1);
#pragma unroll
  for (int off = 16; off; off >>= 1) m = fmaxf(m, __shfl_xor(m, off, 32));
  float s = expf(v0 - m) + ((lane < 8) ? expf(v1 - m) : 0.0f);
#pragma unroll
  for (int off = 16; off; off >>= 1) s += __shfl_xor(s, off, 32);
  float lse = m + logf(s);
  float* lr = LS + (size_t)row * NCLS;
  lr[lane] = v0 - lse;
  if (lane < 8) lr[32 + lane] = v1 - lse;
}

extern "C" void kernel_launch(void* const* d_in, const int* in_sizes, int n_in,
                              void* d_out, int out_size, void* d_ws, size_t ws_size,
                              hipStream_t stream) {
  const float* x   = (const float*)d_in[0];
  const int*   ei  = (const int*)  d_in[1];   // int32 (JAX default x64-off)
  const float* W1  = (const float*)d_in[2];
  const float* b1  = (const float*)d_in[3];
  const float* W2  = (const float*)d_in[4];
  const float* b2  = (const float*)d_in[5];

  const unsigned N = (unsigned)(in_sizes[0] / F_IN);   // 100000
  const unsigned E = (unsigned)(in_sizes[1] / 2);      // 3200000
  const int* src = ei;
  const int* dst = ei + E;

  // workspace layout (floats): [dis | agg1 | h_lin | z_lin]
  float* ws    = (float*)d_ws;
  size_t nA    = ((size_t)N + 63) & ~(size_t)63;
  float* dis   = ws;
  float* agg1  = ws + nA;
  float* hlin  = agg1 + (size_t)N * HID;
  float* zlin  = hlin + (size_t)N * HID;

  float* lsm_out = (float*)d_out;                 // [N x 40] log_softmax
  float* xo_out  = lsm_out + (size_t)N * NCLS;    // [N x 40] x_out (also agg2 accumulator)

  // zero: dis + agg1 (contiguous), and x_out accumulator
  hipMemsetAsync(ws, 0, (nA + (size_t)N * HID) * sizeof(float), stream);
  hipMemsetAsync(xo_out, 0, (size_t)N * NCLS * sizeof(float), stream);

  const unsigned mtiles = N / 16;                 // 6250 exact tiles
  const unsigned gblk   = (mtiles + 3) / 4;       // 4 waves per block

  k_deg<<<(E + 255) / 256, 256, 0, stream>>>(dst, dis, E);
  k_dis<<<(N + 255) / 256, 256, 0, stream>>>(dis, N);

  k_gemm1<<<gblk, 128, 0, stream>>>(x, W1, hlin, (int)mtiles);

  unsigned tot1 = E * HID;                        // 204.8M (edge, feat) atomics, L2-resident
  k_scatter1<<<(tot1 + 255) / 256, 256, 0, stream>>>(src, dst, dis, hlin, agg1, tot1);
  unsigned totf1 = N * HID;
  k_fin1<<<(totf1 + 255) / 256, 256, 0, stream>>>(agg1, hlin, dis, b1, totf1);

  k_gemm2<<<gblk, 128, 0, stream>>>(agg1, W2, zlin, (int)mtiles);

  unsigned tot2 = E * NCLS;                       // 128M
  k_scatter2<<<(tot2 + 255) / 256, 256, 0, stream>>>(src, dst, dis, zlin, xo_out, tot2);
  unsigned totf2 = N * NCLS;
  k_fin2<<<(totf2 + 255) / 256, 256, 0, stream>>>(xo_out, zlin, dis, b2, totf2);

  k_lsm<<<(N + 7) / 8, 256, 0, stream>>>(xo_out, lsm_out, N);
}